// StoxLSTMChannel_80410377716156
// MI455X (gfx1250) — compile-verified
//
#include <hip/hip_runtime.h>
#include <cstdint>

typedef __attribute__((ext_vector_type(2))) float v2f;
typedef __attribute__((ext_vector_type(8))) float v8f;

#define B_  256
#define TH_ 512
#define TA_ 576
#define P_  16
#define D_  512
#define L_  128
#define H_  256

// ---------------------------------------------------------------------------
// WMMA helpers: fp32 16x16x4 (native fp32 matrix path on CDNA5)
// A 16x4: lanes 0-15 row M=lane, VGPR0/1 = K,K+1; lanes 16-31 same rows, K+2,K+3
// B 4x16: lanes 0-15 col N=lane row K/K+1; lanes 16-31 rows K+2/K+3
// C/D 16x16: VGPR v holds row v (lanes 0-15) / row v+8 (lanes 16-31), col=lane&15
// ---------------------------------------------------------------------------
__device__ __forceinline__ v8f wmma4(v2f a, v2f b, v8f c) {
  return __builtin_amdgcn_wmma_f32_16x16x4_f32(false, a, false, b, (short)0, c,
                                               false, false);
}

__device__ __forceinline__ v8f zero8() {
  v8f z;
#pragma unroll
  for (int i = 0; i < 8; ++i) z[i] = 0.0f;
  return z;
}

// A fragment from row-major [16, K] tile (rows base..base+15, cols k..k+3)
__device__ __forceinline__ v2f ldA(const float* base, int lda, int k, int lane) {
  int row = lane & 15;
  int ks  = (lane & 16) ? 2 : 0;
  const float* p = base + (size_t)row * lda + k + ks;
  v2f r; r.x = p[0]; r.y = p[1];
  return r;
}

// B fragment from PACKED weights: Wp[k4][nt][lane][2] -> one coalesced b64/lane
__device__ __forceinline__ v2f ldBpk(const float* __restrict__ Wp, int ntiles,
                                     int k4, int nt, int lane) {
  const float* p = Wp + (((size_t)k4 * ntiles + nt) << 6) + lane * 2;
  v2f r; r.x = p[0]; r.y = p[1];
  return r;
}

__device__ __forceinline__ float sigm(float x)      { return 1.0f / (1.0f + __expf(-x)); }
__device__ __forceinline__ float softplus_(float x) { return (x > 20.0f) ? x : log1pf(__expf(x)); }
__device__ __forceinline__ float gelu_(float x)     { return 0.5f * x * (1.0f + erff(x * 0.70710678118f)); }

// ---------------------------------------------------------------------------
// Fold embedding into input-gate weights: Wx[16,Ng] = W_embed @ Wih,
// bx[Ng] = b_embed @ Wih + bih.  (Tiny; runs once.)
// ---------------------------------------------------------------------------
__global__ void fuse_weights(const float* __restrict__ We, const float* __restrict__ be,
                             const float* __restrict__ Wih, const float* __restrict__ bih,
                             float* __restrict__ Wx, float* __restrict__ bx, int Ng) {
  int n = blockIdx.x * blockDim.x + threadIdx.x;
  if (n >= Ng) return;
#pragma unroll 1
  for (int p = 0; p < P_; ++p) {
    float s = 0.0f;
    for (int d = 0; d < D_; ++d) s += We[p * D_ + d] * Wih[(size_t)d * Ng + n];
    Wx[p * Ng + n] = s;
  }
  float sb = bih[n];
  for (int d = 0; d < D_; ++d) sb += be[d] * Wih[(size_t)d * Ng + n];
  bx[n] = sb;
}

// ---------------------------------------------------------------------------
// Repack row-major W[K,N] into WMMA-B fragment order [K/4][N/16][lane][2] so
// the hot loops issue one contiguous global_load_b64 per lane per fragment.
// ---------------------------------------------------------------------------
__global__ void pack_b(const float* __restrict__ W, float* __restrict__ Wp,
                       int K, int N) {
  int idx = blockIdx.x * blockDim.x + threadIdx.x;
  int total = (K >> 2) * (N >> 4) * 32;
  if (idx >= total) return;
  int lane = idx & 31;
  int nt   = (idx >> 5) % (N >> 4);
  int k4   = idx / (32 * (N >> 4));
  int ks   = (lane & 16) ? 2 : 0;
  int col  = nt * 16 + (lane & 15);
  int k    = k4 * 4 + ks;
  Wp[(size_t)idx * 2 + 0] = W[(size_t)k * N + col];
  Wp[(size_t)idx * 2 + 1] = W[(size_t)(k + 1) * N + col];
}

// ---------------------------------------------------------------------------
// LSTM body: 16 waves, wave owns 16 gate columns (n0 = wave*16 within H=256).
// h lives in LDS, c in VGPRs. Per step: g = p_t@Wx (K=16) + h@Whh (K=256).
// ---------------------------------------------------------------------------
__device__ __forceinline__ void lstm_body(
    const float* __restrict__ patches, const float* __restrict__ WxP,
    const float* __restrict__ bx, const float* __restrict__ WhP,
    const float* __restrict__ bhh, float* __restrict__ hout,
    int T, int col_off, int rev, int b0, float* hsh) {
  const int lane = threadIdx.x & 31;
  const int wave = threadIdx.x >> 5;   // 0..15
  const int NT   = (4 * H_) >> 4;      // 64 column tiles

  for (int i = threadIdx.x; i < 16 * H_; i += 512) hsh[i] = 0.0f;
  v8f c = zero8();
  __syncthreads();

  for (int tt = 0; tt < T; ++tt) {
    const int t = rev ? (T - 1 - tt) : tt;
    v8f acc[4];
#pragma unroll
    for (int g = 0; g < 4; ++g) acc[g] = zero8();

    const float* A = patches + ((size_t)b0 * T + t) * P_;
#pragma unroll
    for (int k = 0; k < P_; k += 4) {
      v2f a = ldA(A, T * P_, k, lane);
#pragma unroll
      for (int g = 0; g < 4; ++g)
        acc[g] = wmma4(a, ldBpk(WxP, NT, k >> 2, g * 16 + wave, lane), acc[g]);
    }
#pragma unroll 2
    for (int k = 0; k < H_; k += 4) {
      if ((k & 63) == 0)  // prefetch next packed K-panel of this wave's stream
        __builtin_prefetch(WhP + (((size_t)(k >> 2) + 16) * NT + wave) * 64, 0, 1);
      v2f a = ldA(hsh, H_, k, lane);
#pragma unroll
      for (int g = 0; g < 4; ++g)
        acc[g] = wmma4(a, ldBpk(WhP, NT, k >> 2, g * 16 + wave, lane), acc[g]);
    }
    __syncthreads();  // everyone done reading hsh

    const int col = lane & 15;
    const int rb  = (lane & 16) ? 8 : 0;
    const int n   = wave * 16 + col;
    const float bi = bx[0 * H_ + n] + bhh[0 * H_ + n];
    const float bf = bx[1 * H_ + n] + bhh[1 * H_ + n];
    const float bg = bx[2 * H_ + n] + bhh[2 * H_ + n];
    const float bo = bx[3 * H_ + n] + bhh[3 * H_ + n];
#pragma unroll
    for (int v = 0; v < 8; ++v) {
      const float gi = sigm(acc[0][v] + bi);
      const float gf = sigm(acc[1][v] + bf);
      const float gg = tanhf(acc[2][v] + bg);
      const float go = sigm(acc[3][v] + bo);
      const float cn = gf * c[v] + gi * gg;
      const float hn = go * tanhf(cn);
      c[v] = cn;
      const int r = rb + v;
      hsh[r * H_ + n] = hn;
      hout[((size_t)(b0 + r) * T + t) * 512 + col_off + n] = hn;
    }
    __syncthreads();  // writes visible before next step's reads
  }
}

// ---------------------------------------------------------------------------
// GRU body: 16 waves, wave owns 32 cols of D=512 (2 subtiles). n-gate keeps
// x-part and h-part separate (r scales only the h-part + bhh).
// ---------------------------------------------------------------------------
__device__ __forceinline__ void gru_body(
    const float* __restrict__ patches, const float* __restrict__ WxP,
    const float* __restrict__ bx, const float* __restrict__ WhP,
    const float* __restrict__ bhh, float* __restrict__ hout,
    int b0, float* hsh) {
  const int lane = threadIdx.x & 31;
  const int wave = threadIdx.x >> 5;
  const int n0   = wave * 32;
  const int NT   = (3 * D_) >> 4;  // 96 column tiles

  for (int i = threadIdx.x; i < 16 * D_; i += 512) hsh[i] = 0.0f;
  __syncthreads();

  for (int t = 0; t < TH_; ++t) {
    v8f rA[2], zA[2], xn[2], hn[2];
#pragma unroll
    for (int s = 0; s < 2; ++s) { rA[s] = zero8(); zA[s] = zero8(); xn[s] = zero8(); hn[s] = zero8(); }

    const float* A = patches + ((size_t)b0 * TH_ + t) * P_;
#pragma unroll
    for (int k = 0; k < P_; k += 4) {
      v2f a = ldA(A, TH_ * P_, k, lane);
#pragma unroll
      for (int s = 0; s < 2; ++s) {
        const int nt = wave * 2 + s;
        rA[s] = wmma4(a, ldBpk(WxP, NT, k >> 2, 0 * 32 + nt, lane), rA[s]);
        zA[s] = wmma4(a, ldBpk(WxP, NT, k >> 2, 1 * 32 + nt, lane), zA[s]);
        xn[s] = wmma4(a, ldBpk(WxP, NT, k >> 2, 2 * 32 + nt, lane), xn[s]);
      }
    }
#pragma unroll 2
    for (int k = 0; k < D_; k += 4) {
      if ((k & 63) == 0)
        __builtin_prefetch(WhP + (((size_t)(k >> 2) + 16) * NT + wave * 2) * 64, 0, 1);
      v2f a = ldA(hsh, D_, k, lane);
#pragma unroll
      for (int s = 0; s < 2; ++s) {
        const int nt = wave * 2 + s;
        rA[s] = wmma4(a, ldBpk(WhP, NT, k >> 2, 0 * 32 + nt, lane), rA[s]);
        zA[s] = wmma4(a, ldBpk(WhP, NT, k >> 2, 1 * 32 + nt, lane), zA[s]);
        hn[s] = wmma4(a, ldBpk(WhP, NT, k >> 2, 2 * 32 + nt, lane), hn[s]);
      }
    }
    __syncthreads();

    const int col = lane & 15;
    const int rb  = (lane & 16) ? 8 : 0;
#pragma unroll
    for (int s = 0; s < 2; ++s) {
      const int n = n0 + s * 16 + col;
      const float br  = bx[n] + bhh[n];
      const float bz  = bx[D_ + n] + bhh[D_ + n];
      const float bxn = bx[2 * D_ + n];
      const float bhn = bhh[2 * D_ + n];
#pragma unroll
      for (int v = 0; v < 8; ++v) {
        const int r = rb + v;
        const float hp = hsh[r * D_ + n];  // own column only: no cross-wave race
        const float rg = sigm(rA[s][v] + br);
        const float zg = sigm(zA[s][v] + bz);
        const float ng = tanhf(xn[s][v] + bxn + rg * (hn[s][v] + bhn));
        const float h  = (1.0f - zg) * ng + zg * hp;
        hsh[r * D_ + n] = h;
        hout[((size_t)(b0 + r) * TH_ + t) * D_ + n] = h;
      }
    }
    __syncthreads();
  }
}

// ---------------------------------------------------------------------------
// Unified recurrence kernel: 48 blocks run fwd-LSTM / bwd-LSTM / GRU
// concurrently (they are independent; the stream would otherwise serialize
// the three latency-bound scans).
// ---------------------------------------------------------------------------
__global__ __launch_bounds__(512) void recurrences_kernel(
    const float* __restrict__ patches_all, const float* __restrict__ patches_hist,
    const float* __restrict__ WfxP, const float* __restrict__ bfx,
    const float* __restrict__ WhfP, const float* __restrict__ bhhf,
    const float* __restrict__ WbxP, const float* __restrict__ bbx,
    const float* __restrict__ WhbP, const float* __restrict__ bhhb,
    const float* __restrict__ WgxP, const float* __restrict__ bgx,
    const float* __restrict__ WhgP, const float* __restrict__ bhhg,
    float* __restrict__ hbi, float* __restrict__ h_seq) {
  __shared__ float hsh[16 * D_];
  const int role = blockIdx.x >> 4;
  const int b0   = (blockIdx.x & 15) * 16;
  if (role == 0)
    lstm_body(patches_all, WfxP, bfx, WhfP, bhhf, hbi, TA_, 0, 0, b0, hsh);
  else if (role == 1)
    lstm_body(patches_all, WbxP, bbx, WhbP, bhhb, hbi, TA_, H_, 1, b0, hsh);
  else
    gru_body(patches_hist, WgxP, bgx, WhgP, bhhg, h_seq, b0, hsh);
}

// ---------------------------------------------------------------------------
// Generic tiled WMMA GEMM with packed B: C[M,N] = act(A@W [+ A2@W2] + bias)
// grid = (N/16, M/128); 8 waves per WG, one 16x16 tile per wave.
// act: 0=none, 1=softplus+1e-5, 2=exact gelu
// ---------------------------------------------------------------------------
__global__ __launch_bounds__(256) void gemm_kernel(
    const float* __restrict__ A, int K, const float* __restrict__ Wp,
    const float* __restrict__ A2, int K2, const float* __restrict__ W2p,
    const float* __restrict__ bias, float* __restrict__ C, int N, int act) {
  const int lane = threadIdx.x & 31;
  const int wave = threadIdx.x >> 5;
  const int nt   = blockIdx.x;
  const int NT   = N >> 4;
  const size_t m0 = ((size_t)blockIdx.y * 8 + wave) * 16;

  v8f acc = zero8();
  const float* Ab = A + m0 * K;
#pragma unroll 2
  for (int k = 0; k < K; k += 4)
    acc = wmma4(ldA(Ab, K, k, lane), ldBpk(Wp, NT, k >> 2, nt, lane), acc);
  if (A2) {  // uniform branch -> EXEC stays all-ones around WMMA
    const float* A2b = A2 + m0 * K2;
#pragma unroll 2
    for (int k = 0; k < K2; k += 4)
      acc = wmma4(ldA(A2b, K2, k, lane), ldBpk(W2p, NT, k >> 2, nt, lane), acc);
  }
  const int col = lane & 15;
  const int rb  = (lane & 16) ? 8 : 0;
  const float bv = bias[nt * 16 + col];
#pragma unroll
  for (int v = 0; v < 8; ++v) {
    float x = acc[v] + bv;
    if (act == 1)      x = softplus_(x) + 1e-5f;
    else if (act == 2) x = gelu_(x);
    C[(m0 + rb + v) * (size_t)N + nt * 16 + col] = x;
  }
}

// z = qm[:, :TH] + qs[:, :TH] * eps
__global__ void z_kernel(const float* __restrict__ qm, const float* __restrict__ qs,
                         const float* __restrict__ eps, float* __restrict__ z, int n) {
  int i = blockIdx.x * blockDim.x + threadIdx.x;
  if (i >= n) return;
  int l = i % L_;
  int t = (i / L_) % TH_;
  int b = i / (L_ * TH_);
  size_t q = ((size_t)b * TA_ + t) * L_ + l;
  z[i] = qm[q] + qs[q] * eps[i];
}

// y = e2 @ W_head + b_head  (N=1, dot per row)
__global__ void head_kernel(const float* __restrict__ e2, const float* __restrict__ Wh,
                            const float* __restrict__ bh, float* __restrict__ y, int n) {
  int i = blockIdx.x * blockDim.x + threadIdx.x;
  if (i >= n) return;
  const float* row = e2 + (size_t)i * D_;
  float s = bh[0];
#pragma unroll 4
  for (int d = 0; d < D_; ++d) s += row[d] * Wh[d];
  y[i] = s;
}

// ---------------------------------------------------------------------------
extern "C" void kernel_launch(void* const* d_in, const int* in_sizes, int n_in,
                              void* d_out, int out_size, void* d_ws, size_t ws_size,
                              hipStream_t stream) {
  const float* patches_hist = (const float*)d_in[0];
  const float* patches_all  = (const float*)d_in[1];
  const float* eps     = (const float*)d_in[2];
  const float* W_embed = (const float*)d_in[3];
  const float* b_embed = (const float*)d_in[4];
  const float* Wih_f = (const float*)d_in[5];
  const float* Whh_f = (const float*)d_in[6];
  const float* bih_f = (const float*)d_in[7];
  const float* bhh_f = (const float*)d_in[8];
  const float* Wih_b = (const float*)d_in[9];
  const float* Whh_b = (const float*)d_in[10];
  const float* bih_b = (const float*)d_in[11];
  const float* bhh_b = (const float*)d_in[12];
  const float* Wqm = (const float*)d_in[13];
  const float* bqm = (const float*)d_in[14];
  const float* Wqs = (const float*)d_in[15];
  const float* bqs = (const float*)d_in[16];
  const float* Wih_g = (const float*)d_in[17];
  const float* Whh_g = (const float*)d_in[18];
  const float* bih_g = (const float*)d_in[19];
  const float* bhh_g = (const float*)d_in[20];
  const float* Wpm = (const float*)d_in[21];
  const float* bpm = (const float*)d_in[22];
  const float* Wps = (const float*)d_in[23];
  const float* bps = (const float*)d_in[24];
  const float* W_e1 = (const float*)d_in[25];
  const float* b_e1 = (const float*)d_in[26];
  const float* W_e2 = (const float*)d_in[27];
  const float* b_e2 = (const float*)d_in[28];
  const float* W_head = (const float*)d_in[29];
  const float* b_head = (const float*)d_in[30];

  // outputs concatenated: y_hist, prior_m, prior_s, qm_all, qs_all, h_seq
  float* out     = (float*)d_out;
  float* y_hist  = out;
  float* prior_m = y_hist + (size_t)B_ * TH_;
  float* prior_s = prior_m + (size_t)B_ * TH_ * L_;
  float* qm_all  = prior_s + (size_t)B_ * TH_ * L_;
  float* qs_all  = qm_all + (size_t)B_ * TA_ * L_;
  float* h_seq   = qs_all + (size_t)B_ * TA_ * L_;

  // workspace layout (e2 aliases hbi: hbi is dead after the posterior heads)
  float* ws   = (float*)d_ws;
  float* hbi  = ws;                                 // B*TA*512
  float* zbuf = hbi + (size_t)B_ * TA_ * 512;       // B*TH*L
  float* e1   = zbuf + (size_t)B_ * TH_ * L_;       // B*TH*512
  float* p    = e1 + (size_t)B_ * TH_ * 512;
  // row-major fused input-gate weights (temp) + biases
  float* Wfx = p;            p += 16 * 1024;
  float* bfx = p;            p += 1024;
  float* Wbx = p;            p += 16 * 1024;
  float* bbx = p;            p += 1024;
  float* Wgx = p;            p += 16 * 1536;
  float* bgx = p;            p += 1536;
  // packed (fragment-order) weights
  float* WfxP  = p;          p += 16 * 1024;        // K=16
  float* WbxP  = p;          p += 16 * 1024;
  float* WgxP  = p;          p += 16 * 1536;
  float* WhfP  = p;          p += 256 * 1024;
  float* WhbP  = p;          p += 256 * 1024;
  float* WhgP  = p;          p += 512 * 1536;
  float* WqmP  = p;          p += 512 * 128;
  float* WqsP  = p;          p += 512 * 128;
  float* WpmP  = p;          p += 512 * 128;
  float* WpsP  = p;          p += 512 * 128;
  float* We1aP = p;          p += 512 * 512;
  float* We1bP = p;          p += 128 * 512;
  float* We2P  = p;          p += 512 * 512;
  float* e2 = hbi;  // alias

  // 1) fold embedding into input-gate weights, then repack everything
  fuse_weights<<<dim3(4), 256, 0, stream>>>(W_embed, b_embed, Wih_f, bih_f, Wfx, bfx, 4 * H_);
  fuse_weights<<<dim3(4), 256, 0, stream>>>(W_embed, b_embed, Wih_b, bih_b, Wbx, bbx, 4 * H_);
  fuse_weights<<<dim3(6), 256, 0, stream>>>(W_embed, b_embed, Wih_g, bih_g, Wgx, bgx, 3 * D_);

  auto packN = [](int K, int N) { return ((K >> 2) * (N >> 4) * 32 + 255) / 256; };
  pack_b<<<dim3(packN(16, 1024)), 256, 0, stream>>>(Wfx, WfxP, 16, 1024);
  pack_b<<<dim3(packN(16, 1024)), 256, 0, stream>>>(Wbx, WbxP, 16, 1024);
  pack_b<<<dim3(packN(16, 1536)), 256, 0, stream>>>(Wgx, WgxP, 16, 1536);
  pack_b<<<dim3(packN(256, 1024)), 256, 0, stream>>>(Whh_f, WhfP, 256, 1024);
  pack_b<<<dim3(packN(256, 1024)), 256, 0, stream>>>(Whh_b, WhbP, 256, 1024);
  pack_b<<<dim3(packN(512, 1536)), 256, 0, stream>>>(Whh_g, WhgP, 512, 1536);
  pack_b<<<dim3(packN(512, 128)), 256, 0, stream>>>(Wqm, WqmP, 512, 128);
  pack_b<<<dim3(packN(512, 128)), 256, 0, stream>>>(Wqs, WqsP, 512, 128);
  pack_b<<<dim3(packN(512, 128)), 256, 0, stream>>>(Wpm, WpmP, 512, 128);
  pack_b<<<dim3(packN(512, 128)), 256, 0, stream>>>(Wps, WpsP, 512, 128);
  pack_b<<<dim3(packN(512, 512)), 256, 0, stream>>>(W_e1, We1aP, 512, 512);
  pack_b<<<dim3(packN(128, 512)), 256, 0, stream>>>(W_e1 + (size_t)D_ * D_, We1bP, 128, 512);
  pack_b<<<dim3(packN(512, 512)), 256, 0, stream>>>(W_e2, We2P, 512, 512);

  // 2) fwd-LSTM + bwd-LSTM + GRU, all concurrent (48 WGs)
  recurrences_kernel<<<dim3(48), 512, 0, stream>>>(
      patches_all, patches_hist,
      WfxP, bfx, WhfP, bhh_f,
      WbxP, bbx, WhbP, bhh_b,
      WgxP, bgx, WhgP, bhh_g,
      hbi, h_seq);

  // 3) posterior heads
  gemm_kernel<<<dim3(L_ / 16, (B_ * TA_) / 128), 256, 0, stream>>>(
      hbi, 512, WqmP, nullptr, 0, nullptr, bqm, qm_all, L_, 0);
  gemm_kernel<<<dim3(L_ / 16, (B_ * TA_) / 128), 256, 0, stream>>>(
      hbi, 512, WqsP, nullptr, 0, nullptr, bqs, qs_all, L_, 1);

  // 4) prior heads
  gemm_kernel<<<dim3(L_ / 16, (B_ * TH_) / 128), 256, 0, stream>>>(
      h_seq, D_, WpmP, nullptr, 0, nullptr, bpm, prior_m, L_, 0);
  gemm_kernel<<<dim3(L_ / 16, (B_ * TH_) / 128), 256, 0, stream>>>(
      h_seq, D_, WpsP, nullptr, 0, nullptr, bps, prior_s, L_, 1);

  // 5) reparameterized sample
  const int nz = B_ * TH_ * L_;
  z_kernel<<<dim3(nz / 256), 256, 0, stream>>>(qm_all, qs_all, eps, zbuf, nz);

  // 6) emission MLP: concat handled as two K-panels of W_e1
  gemm_kernel<<<dim3(D_ / 16, (B_ * TH_) / 128), 256, 0, stream>>>(
      h_seq, D_, We1aP, zbuf, L_, We1bP, b_e1, e1, D_, 2);
  gemm_kernel<<<dim3(D_ / 16, (B_ * TH_) / 128), 256, 0, stream>>>(
      e1, D_, We2P, nullptr, 0, nullptr, b_e2, e2, D_, 0);
  head_kernel<<<dim3((B_ * TH_) / 256), 256, 0, stream>>>(e2, W_head, b_head, y_hist, B_ * TH_);
}